// ProductManifoldTransformerLayer_43052752175141
// MI455X (gfx1250) — compile-verified
//
#include <hip/hip_runtime.h>
#include <math.h>

#define EPS_C 1e-8f
#define INV_SQRT_C 0.999999995f

typedef __attribute__((ext_vector_type(2))) float v2f;
typedef __attribute__((ext_vector_type(8))) float v8f;

__device__ __forceinline__ v8f wmma4(v2f a, v2f b, v8f c) {
  // V_WMMA_F32_16X16X4_F32: D = A(16x4 f32) x B(4x16 f32) + C(16x16 f32)
  return __builtin_amdgcn_wmma_f32_16x16x4_f32(
      false, a, false, b, (short)0, c, false, false);
}

// ---------------- LayerNorm: one wave (32 lanes) per row of D=512 ----------
__global__ void ln_kernel(const float* __restrict__ x,
                          const float* __restrict__ w,
                          const float* __restrict__ b,
                          float* __restrict__ out, int D) {
  const int row = blockIdx.x;
  const int lane = threadIdx.x;
  const float* xr = x + (size_t)row * D;
  float vals[16];
  float s = 0.f;
  const int per = D / 32;  // 16
  for (int i = 0; i < per; ++i) {
    vals[i] = xr[lane + i * 32];
    s += vals[i];
  }
  for (int off = 16; off >= 1; off >>= 1) s += __shfl_xor(s, off, 32);
  const float mean = s / (float)D;
  float vs = 0.f;
  for (int i = 0; i < per; ++i) {
    float d = vals[i] - mean;
    vs += d * d;
  }
  for (int off = 16; off >= 1; off >>= 1) vs += __shfl_xor(vs, off, 32);
  const float rstd = rsqrtf(vs / (float)D + 1e-5f);
  float* orow = out + (size_t)row * D;
  for (int i = 0; i < per; ++i) {
    int c = lane + i * 32;
    orow[c] = (vals[i] - mean) * rstd * w[c] + b[c];
  }
}

// ---------------- Fused QKV projection (per-head 64x64 weights) ------------
// grid: (64/16, S/16, B*H), block: 32 (one wave per 16x16 tile, 3 accums)
__global__ void qkv_kernel(const float* __restrict__ h,
                           const float* __restrict__ Wq,
                           const float* __restrict__ Wk,
                           const float* __restrict__ Wv,
                           float* __restrict__ q, float* __restrict__ k,
                           float* __restrict__ v, int S, int D) {
  const int bh = blockIdx.z;
  const int b = bh >> 3, hd = bh & 7;
  const float* A = h + (size_t)b * S * D + hd * 64;  // row stride D
  const float* wq = Wq + (size_t)hd * 64 * 64;       // [e][d], ld 64
  const float* wk = Wk + (size_t)hd * 64 * 64;
  const float* wv = Wv + (size_t)hd * 64 * 64;
  const int n0 = blockIdx.x * 16, m0 = blockIdx.y * 16;
  const int lane = threadIdx.x & 31;
  const int half = lane >> 4, lm = lane & 15;

  v8f cq = {}, ck = {}, cv = {};
  for (int k0 = 0; k0 < 64; k0 += 4) {
    v2f a = *(const v2f*)(A + (size_t)(m0 + lm) * D + k0 + half * 2);
    v2f bq = *(const v2f*)(wq + (size_t)(n0 + lm) * 64 + k0 + half * 2);
    v2f bk = *(const v2f*)(wk + (size_t)(n0 + lm) * 64 + k0 + half * 2);
    v2f bv = *(const v2f*)(wv + (size_t)(n0 + lm) * 64 + k0 + half * 2);
    cq = wmma4(a, bq, cq);
    ck = wmma4(a, bk, ck);
    cv = wmma4(a, bv, cv);
  }
  float* qo = q + (size_t)bh * S * 64;
  float* ko = k + (size_t)bh * S * 64;
  float* vo = v + (size_t)bh * S * 64;
  for (int r = 0; r < 8; ++r) {
    size_t idx = (size_t)(m0 + half * 8 + r) * 64 + n0 + lm;
    qo[idx] = cq[r];
    ko[idx] = ck[r];
    vo[idx] = cv[r];
  }
}

// ---------------- squared norms over head dim (64) -------------------------
__global__ void sumsq_kernel(const float* __restrict__ q,
                             const float* __restrict__ k,
                             float* __restrict__ qsq, float* __restrict__ ksq,
                             int n) {
  int i = blockIdx.x * blockDim.x + threadIdx.x;
  if (i >= n) return;
  const float* qr = q + (size_t)i * 64;
  const float* kr = k + (size_t)i * 64;
  float s = 0.f, t = 0.f;
  for (int j = 0; j < 64; ++j) {
    s += qr[j] * qr[j];
    t += kr[j] * kr[j];
  }
  qsq[i] = s;
  ksq[i] = t;
}

// ---------------- product-manifold flash attention -------------------------
// grid: (S/16, B*H), block 32. One wave owns a 16-row query tile.
__global__ void attn_kernel(const float* __restrict__ q,
                            const float* __restrict__ k,
                            const float* __restrict__ v,
                            const float* __restrict__ qsq,
                            const float* __restrict__ ksq,
                            const float* __restrict__ gw,
                            const float* __restrict__ temp,
                            float* __restrict__ o, int S) {
  const int bh = blockIdx.y;
  const int hd = bh & 7;
  const int s0 = blockIdx.x * 16;
  const float* qb = q + (size_t)bh * S * 64;
  const float* kb = k + (size_t)bh * S * 64;
  const float* vb = v + (size_t)bh * S * 64;
  const float* qsb = qsq + (size_t)bh * S;
  const float* ksb = ksq + (size_t)bh * S;
  const int lane = threadIdx.x & 31;
  const int half = lane >> 4, lm = lane & 15;

  // per-head geometry mixing weights: softmax over 3
  float g0 = gw[hd * 3 + 0], g1 = gw[hd * 3 + 1], g2 = gw[hd * 3 + 2];
  float gm = fmaxf(g0, fmaxf(g1, g2));
  float e0 = expf(g0 - gm), e1 = expf(g1 - gm), e2 = expf(g2 - gm);
  float esum = e0 + e1 + e2;
  const float w0 = e0 / esum, w1 = e1 / esum, w2 = e2 / esum;
  const float tmp = temp[hd];

  // preload 16 A-fragments of the query tile (K = 64)
  v2f qa[16];
  for (int kk = 0; kk < 16; ++kk)
    qa[kk] = *(const v2f*)(qb + (size_t)(s0 + lm) * 64 + kk * 4 + half * 2);

  // per-lane row stats (rows m = half*8 + r)
  float qs[8], qn_[8], mrow[8], lrow[8];
  for (int r = 0; r < 8; ++r) {
    float t = qsb[s0 + half * 8 + r];
    qs[r] = t;
    qn_[r] = sqrtf(t);
    mrow[r] = -INFINITY;
    lrow[r] = 0.f;
  }
  v8f acc[4] = {{}, {}, {}, {}};

  __shared__ float plds[16 * 16];

  for (int t0 = 0; t0 < S; t0 += 16) {
    // S = q . k^T over 16x16 tile
    v8f sfrag = {};
    for (int kk = 0; kk < 16; ++kk) {
      v2f bb = *(const v2f*)(kb + (size_t)(t0 + lm) * 64 + kk * 4 + half * 2);
      sfrag = wmma4(qa[kk], bb, sfrag);
    }
    const float kscol = ksb[t0 + lm];
    const float kncol = sqrtf(kscol);

    float p[8], nmax[8];
    for (int r = 0; r < 8; ++r) {
      float qk = sfrag[r];
      float dsq = fmaxf(qs[r] + kscol - 2.f * qk, 0.f);
      float euc = sqrtf(dsq + EPS_C);
      float denom = (1.f - qs[r]) * (1.f - kscol) + EPS_C;
      float ca = 1.f + 2.f * dsq / denom;
      ca = fminf(fmaxf(ca, 1.f), 1e6f);
      float hyp = logf(ca + sqrtf(ca * ca - 1.f)) * INV_SQRT_C;
      float cosn = qk / ((qn_[r] + EPS_C) * (kncol + EPS_C));
      cosn = fminf(fmaxf(cosn, -1.f + EPS_C), 1.f - EPS_C);
      float sph = acosf(cosn) * INV_SQRT_C;
      float sim = -(w0 * hyp + w1 * euc + w2 * sph) * tmp;
      // row max across the 16 lanes of this half
      float rm = sim;
      rm = fmaxf(rm, __shfl_xor(rm, 1, 32));
      rm = fmaxf(rm, __shfl_xor(rm, 2, 32));
      rm = fmaxf(rm, __shfl_xor(rm, 4, 32));
      rm = fmaxf(rm, __shfl_xor(rm, 8, 32));
      nmax[r] = fmaxf(mrow[r], rm);
      p[r] = expf(sim - nmax[r]);
    }
    for (int r = 0; r < 8; ++r) {
      float ps = p[r];
      ps += __shfl_xor(ps, 1, 32);
      ps += __shfl_xor(ps, 2, 32);
      ps += __shfl_xor(ps, 4, 32);
      ps += __shfl_xor(ps, 8, 32);
      float corr = expf(mrow[r] - nmax[r]);
      lrow[r] = lrow[r] * corr + ps;
      mrow[r] = nmax[r];
      acc[0][r] *= corr;
      acc[1][r] *= corr;
      acc[2][r] *= corr;
      acc[3][r] *= corr;
    }
    // C-frag -> A-frag conversion through LDS
    for (int r = 0; r < 8; ++r) plds[(half * 8 + r) * 16 + lm] = p[r];
    __syncthreads();
    for (int kk0 = 0; kk0 < 16; kk0 += 4) {
      v2f pa = *(const v2f*)(&plds[lm * 16 + kk0 + half * 2]);
      for (int e = 0; e < 4; ++e) {
        float b0 = vb[(size_t)(t0 + kk0 + half * 2 + 0) * 64 + e * 16 + lm];
        float b1 = vb[(size_t)(t0 + kk0 + half * 2 + 1) * 64 + e * 16 + lm];
        v2f bb = {b0, b1};
        acc[e] = wmma4(pa, bb, acc[e]);
      }
    }
    __syncthreads();
  }
  float* ob = o + (size_t)bh * S * 64;
  for (int r = 0; r < 8; ++r) {
    float inv = 1.f / lrow[r];
    size_t base = (size_t)(s0 + half * 8 + r) * 64;
    for (int e = 0; e < 4; ++e) ob[base + e * 16 + lm] = acc[e][r] * inv;
  }
}

// ---------------- merge heads: o[B,H,S,64] -> ctx[B,S,D] -------------------
__global__ void merge_kernel(const float* __restrict__ o,
                             float* __restrict__ ctx, int S, int D) {
  int idx = blockIdx.x * blockDim.x + threadIdx.x;
  int b = idx / (S * D);
  int rem = idx - b * S * D;
  int s = rem / D;
  int dc = rem - s * D;
  int hd = dc >> 6, e = dc & 63;
  ctx[idx] = o[(((size_t)(b * 8 + hd) * S) + s) * 64 + e];
}

// ---------------- generic WMMA GEMM: C = A @ W^T + bias [gelu] [+resid] ----
// 16x64 tile per wave: one A fragment feeds 4 WMMAs per K-step.
// grid: (N/64, M/16), block 32
__global__ void gemm_kernel(const float* __restrict__ A, int lda,
                            const float* __restrict__ W, int ldw,
                            const float* __restrict__ bias,
                            const float* __restrict__ resid,
                            float* __restrict__ C, int ldc, int K, int gelu) {
  const int n0 = blockIdx.x * 64, m0 = blockIdx.y * 16;
  const int lane = threadIdx.x & 31;
  const int half = lane >> 4, lm = lane & 15;
  v8f acc[4] = {{}, {}, {}, {}};
  const float* Arow = A + (size_t)(m0 + lm) * lda;
  const float* Wrow0 = W + (size_t)(n0 + lm) * ldw;
  for (int k0 = 0; k0 < K; k0 += 4) {
    v2f a = *(const v2f*)(Arow + k0 + half * 2);
    for (int j = 0; j < 4; ++j) {
      v2f b = *(const v2f*)(Wrow0 + (size_t)j * 16 * ldw + k0 + half * 2);
      acc[j] = wmma4(a, b, acc[j]);
    }
  }
  for (int j = 0; j < 4; ++j) {
    for (int r = 0; r < 8; ++r) {
      int m = m0 + half * 8 + r;
      int n = n0 + j * 16 + lm;
      float val = acc[j][r] + (bias ? bias[n] : 0.f);
      if (gelu) val = 0.5f * val * (1.f + erff(val * 0.70710678118654752f));
      if (resid) val += resid[(size_t)m * ldc + n];
      C[(size_t)m * ldc + n] = val;
    }
  }
}

// ---------------- softmax(gw) tail output ----------------------------------
__global__ void gw_out_kernel(const float* __restrict__ gw,
                              float* __restrict__ out) {
  int hd = threadIdx.x;
  if (hd < 8) {
    float g0 = gw[hd * 3 + 0], g1 = gw[hd * 3 + 1], g2 = gw[hd * 3 + 2];
    float gm = fmaxf(g0, fmaxf(g1, g2));
    float e0 = expf(g0 - gm), e1 = expf(g1 - gm), e2 = expf(g2 - gm);
    float es = e0 + e1 + e2;
    out[hd * 3 + 0] = e0 / es;
    out[hd * 3 + 1] = e1 / es;
    out[hd * 3 + 2] = e2 / es;
  }
}

extern "C" void kernel_launch(void* const* d_in, const int* in_sizes, int n_in,
                              void* d_out, int out_size, void* d_ws,
                              size_t ws_size, hipStream_t stream) {
  const int B = 4, S = 1024, D = 512, H = 8, FF = 2048;
  const size_t BSD = (size_t)B * S * D;       // 2097152
  const size_t BHS = (size_t)B * H * S;       // 32768
  const size_t BSF = (size_t)B * S * FF;      // 8388608

  const float* x = (const float*)d_in[0];
  const float* Wq = (const float*)d_in[1];
  const float* Wk = (const float*)d_in[2];
  const float* Wv = (const float*)d_in[3];
  const float* gw = (const float*)d_in[4];
  const float* temp = (const float*)d_in[5];
  const float* Wo = (const float*)d_in[6];
  const float* bo = (const float*)d_in[7];
  const float* ln1w = (const float*)d_in[8];
  const float* ln1b = (const float*)d_in[9];
  const float* ln2w = (const float*)d_in[10];
  const float* ln2b = (const float*)d_in[11];
  const float* W1 = (const float*)d_in[12];
  const float* b1 = (const float*)d_in[13];
  const float* W2 = (const float*)d_in[14];
  const float* b2 = (const float*)d_in[15];

  float* ws = (float*)d_ws;
  float* h = ws;            ws += BSD;
  float* qb = ws;           ws += BSD;
  float* kb = ws;           ws += BSD;
  float* vb = ws;           ws += BSD;
  float* qsq = ws;          ws += BHS;
  float* ksq = ws;          ws += BHS;
  float* o = ws;            ws += BSD;
  float* ctx = ws;          ws += BSD;
  float* x2 = ws;           ws += BSD;
  float* h2 = ws;           ws += BSD;
  float* ff1 = ws;          ws += BSF;

  float* y = (float*)d_out;

  // 1) LayerNorm1
  ln_kernel<<<dim3(B * S), dim3(32), 0, stream>>>(x, ln1w, ln1b, h, D);
  // 2) fused QKV per head
  qkv_kernel<<<dim3(4, S / 16, B * H), dim3(32), 0, stream>>>(
      h, Wq, Wk, Wv, qb, kb, vb, S, D);
  // 3) squared norms
  sumsq_kernel<<<dim3((int)(BHS / 256)), dim3(256), 0, stream>>>(
      qb, kb, qsq, ksq, (int)BHS);
  // 4) product-manifold flash attention
  attn_kernel<<<dim3(S / 16, B * H), dim3(32), 0, stream>>>(
      qb, kb, vb, qsq, ksq, gw, temp, o, S);
  // 5) merge heads
  merge_kernel<<<dim3((int)(BSD / 256)), dim3(256), 0, stream>>>(o, ctx, S, D);
  // 6) attn_out = ctx @ Wo^T + bo ; x2 = x + attn_out
  gemm_kernel<<<dim3(D / 64, B * S / 16), dim3(32), 0, stream>>>(
      ctx, D, Wo, D, bo, x, x2, D, D, 0);
  // 7) LayerNorm2
  ln_kernel<<<dim3(B * S), dim3(32), 0, stream>>>(x2, ln2w, ln2b, h2, D);
  // 8) ff1 = gelu(h2 @ W1^T + b1)
  gemm_kernel<<<dim3(FF / 64, B * S / 16), dim3(32), 0, stream>>>(
      h2, D, W1, D, b1, nullptr, ff1, FF, D, 1);
  // 9) y = x2 + ff1 @ W2^T + b2
  gemm_kernel<<<dim3(D / 64, B * S / 16), dim3(32), 0, stream>>>(
      ff1, FF, W2, FF, b2, x2, y, D, FF, 0);
  // 10) softmax(gw) tail
  gw_out_kernel<<<dim3(1), dim3(32), 0, stream>>>(gw, y + BSD);
}